// GCNEncoder_584115552795
// MI455X (gfx1250) — compile-verified
//
#include <hip/hip_runtime.h>

// ---------------------------------------------------------------------------
// GCN 2-layer forward for MI455X (gfx1250, wave32).
// Dense projections use V_WMMA_F32_16X16X4_F32 (fp32 matrix cores);
// edge aggregation uses float4 gathers + global_atomic_add_f32 scatters.
// ---------------------------------------------------------------------------

typedef __attribute__((ext_vector_type(2))) float v2f;
typedef __attribute__((ext_vector_type(8))) float v8f;

#define IN_CH  16
#define HIDDEN 128
#define OUT_CH 16

// ---- degree / norm ---------------------------------------------------------

__global__ void k_fill_deg(float* __restrict__ deg, int n) {
  int i = blockIdx.x * blockDim.x + threadIdx.x;
  if (i < n) deg[i] = 1.0f;  // self-loop contributes weight 1
}

__global__ void k_accum_deg(const int* __restrict__ dst, const float* __restrict__ w,
                            float* __restrict__ deg, int e) {
  int i = blockIdx.x * blockDim.x + threadIdx.x;
  if (i < e) atomicAdd(&deg[dst[i]], w[i]);
}

__global__ void k_dinv(const float* __restrict__ deg, float* __restrict__ dinv, int n) {
  int i = blockIdx.x * blockDim.x + threadIdx.x;
  if (i < n) {
    float d = deg[i];
    dinv[i] = (d > 0.0f) ? rsqrtf(d) : 0.0f;
  }
}

// ---- WMMA GEMMs ------------------------------------------------------------
// One wave per 16x16 output tile. A: 16x4 f32 (2 VGPR/lane), B: 4x16 f32,
// C/D: v8f with row = 8*(lane>=16)+j, col = lane&15.

__global__ void k_gemm1(const float* __restrict__ x, const float* __restrict__ W,
                        float* __restrict__ h, int n) {
  int wave = blockIdx.x * (blockDim.x >> 5) + (threadIdx.x >> 5);
  int lane = threadIdx.x & 31;
  int nrt  = n >> 4;                       // row tiles (n is a multiple of 16)
  if (wave >= nrt * (HIDDEN / 16)) return; // wave-uniform guard (EXEC all-1 for WMMA)
  int rt = wave / (HIDDEN / 16);
  int ct = wave % (HIDDEN / 16);
  int half = lane >> 4;
  int l15  = lane & 15;
  const float* xr = x + (size_t)(rt * 16 + l15) * IN_CH;
  v8f acc = {};
#pragma unroll
  for (int k0 = 0; k0 < IN_CH; k0 += 4) {
    int ka = k0 + half * 2;
    v2f a, b;
    a.x = xr[ka];
    a.y = xr[ka + 1];
    b.x = W[(size_t)ka * HIDDEN + ct * 16 + l15];
    b.y = W[(size_t)(ka + 1) * HIDDEN + ct * 16 + l15];
    acc = __builtin_amdgcn_wmma_f32_16x16x4_f32(false, a, false, b, (short)0, acc,
                                                false, false);
  }
#pragma unroll
  for (int j = 0; j < 8; ++j)
    h[(size_t)(rt * 16 + half * 8 + j) * HIDDEN + ct * 16 + l15] = acc[j];
}

__global__ void k_gemm2(const float* __restrict__ h, const float* __restrict__ W,
                        float* __restrict__ y, int n) {
  int wave = blockIdx.x * (blockDim.x >> 5) + (threadIdx.x >> 5);
  int lane = threadIdx.x & 31;
  int nrt  = n >> 4;
  if (wave >= nrt) return;                 // wave-uniform guard
  int half = lane >> 4;
  int l15  = lane & 15;
  const float* hr = h + (size_t)(wave * 16 + l15) * HIDDEN;
  v8f acc = {};
#pragma unroll 8
  for (int k0 = 0; k0 < HIDDEN; k0 += 4) {
    int ka = k0 + half * 2;
    v2f a, b;
    a.x = hr[ka];
    a.y = hr[ka + 1];
    b.x = W[(size_t)ka * OUT_CH + l15];
    b.y = W[(size_t)(ka + 1) * OUT_CH + l15];
    acc = __builtin_amdgcn_wmma_f32_16x16x4_f32(false, a, false, b, (short)0, acc,
                                                false, false);
  }
#pragma unroll
  for (int j = 0; j < 8; ++j)
    y[(size_t)(wave * 16 + half * 8 + j) * OUT_CH + l15] = acc[j];
}

// ---- self-loop init + edge scatter (layer 1, 128 ch) -----------------------

__global__ void k_self_init128(const float* __restrict__ dinv, const float* __restrict__ h1,
                               float* __restrict__ out1, int n) {
  int i = blockIdx.x * blockDim.x + threadIdx.x;
  if (i < n * HIDDEN) {
    float di = dinv[i >> 7];               // norm = dinv*1*dinv for self edge
    out1[i] = di * di * h1[i];
  }
}

// one wave32 per edge: lane t handles channels 4t..4t+3 (float4 gather + 4 atomics)
__global__ void k_scatter128(const int* __restrict__ src, const int* __restrict__ dst,
                             const float* __restrict__ w, const float* __restrict__ dinv,
                             const float* __restrict__ h1, float* __restrict__ out1,
                             int e) {
  int warp = blockIdx.x * (blockDim.x >> 5) + (threadIdx.x >> 5);
  int lane = threadIdx.x & 31;
  if (warp >= e) return;
  int s = src[warp];
  int d = dst[warp];
  float norm = dinv[s] * w[warp] * dinv[d];
  const float4 v = *((const float4*)(h1 + (size_t)s * HIDDEN) + lane);
  float* o = out1 + (size_t)d * HIDDEN + lane * 4;
  atomicAdd(o + 0, v.x * norm);
  atomicAdd(o + 1, v.y * norm);
  atomicAdd(o + 2, v.z * norm);
  atomicAdd(o + 3, v.w * norm);
}

__global__ void k_bias_relu(const float* __restrict__ out1, const float* __restrict__ b1,
                            float* __restrict__ h1b, int n) {
  int i = blockIdx.x * blockDim.x + threadIdx.x;
  if (i < n * HIDDEN) h1b[i] = fmaxf(out1[i] + b1[i & (HIDDEN - 1)], 0.0f);
}

// ---- self-loop init + edge scatter (layer 2, 16 ch) ------------------------

__global__ void k_self_init16(const float* __restrict__ dinv, const float* __restrict__ h2,
                              const float* __restrict__ b2, float* __restrict__ out, int n) {
  int i = blockIdx.x * blockDim.x + threadIdx.x;
  if (i < n * OUT_CH) {
    float di = dinv[i >> 4];
    out[i] = di * di * h2[i] + b2[i & (OUT_CH - 1)];
  }
}

// 4 threads per edge, each covers 4 channels via float4
__global__ void k_scatter16(const int* __restrict__ src, const int* __restrict__ dst,
                            const float* __restrict__ w, const float* __restrict__ dinv,
                            const float* __restrict__ h2, float* __restrict__ out, int e) {
  int tid = blockIdx.x * blockDim.x + threadIdx.x;
  if (tid >= e * 4) return;
  int ed = tid >> 2;
  int g  = tid & 3;
  int s = src[ed];
  int d = dst[ed];
  float norm = dinv[s] * w[ed] * dinv[d];
  const float4 v = *((const float4*)(h2 + (size_t)s * OUT_CH) + g);
  float* o = out + (size_t)d * OUT_CH + g * 4;
  atomicAdd(o + 0, v.x * norm);
  atomicAdd(o + 1, v.y * norm);
  atomicAdd(o + 2, v.z * norm);
  atomicAdd(o + 3, v.w * norm);
}

// ---------------------------------------------------------------------------

extern "C" void kernel_launch(void* const* d_in, const int* in_sizes, int n_in,
                              void* d_out, int out_size, void* d_ws, size_t ws_size,
                              hipStream_t stream) {
  const float* x  = (const float*)d_in[0];
  const int*   ei = (const int*)d_in[1];   // [2, E] flat
  const float* ew = (const float*)d_in[2];
  const float* W1 = (const float*)d_in[3];
  const float* b1 = (const float*)d_in[4];
  const float* W2 = (const float*)d_in[5];
  const float* b2 = (const float*)d_in[6];
  float* out = (float*)d_out;

  const int N = in_sizes[0] / IN_CH;       // 100000 (multiple of 16)
  const int E = in_sizes[2];               // 3200000
  const int* src = ei;
  const int* dst = ei + E;

  // workspace layout (floats): deg[N] | dinv[N] | h1[N*128] | out1[N*128]
  float* deg  = (float*)d_ws;
  float* dinv = deg + (size_t)N;
  float* h1   = dinv + (size_t)N;          // also holds relu(out1+b1)
  float* out1 = h1 + (size_t)N * HIDDEN;   // also holds h2 (N*16)

  const int T = 256;

  // norm coefficients
  k_fill_deg<<<(N + T - 1) / T, T, 0, stream>>>(deg, N);
  k_accum_deg<<<(E + T - 1) / T, T, 0, stream>>>(dst, ew, deg, E);
  k_dinv<<<(N + T - 1) / T, T, 0, stream>>>(deg, dinv, N);

  // layer 1: h1 = x @ W1 ; out1 = scatter(norm * h1) ; h1b = relu(out1 + b1)
  {
    int waves = (N / 16) * (HIDDEN / 16);      // 50000 tiles
    int blocks = (waves + 3) / 4;              // 4 waves (128 thr) per block
    k_gemm1<<<blocks, 128, 0, stream>>>(x, W1, h1, N);
  }
  k_self_init128<<<(N * HIDDEN + T - 1) / T, T, 0, stream>>>(dinv, h1, out1, N);
  k_scatter128<<<(E + 7) / 8, 256, 0, stream>>>(src, dst, ew, dinv, h1, out1, E);
  k_bias_relu<<<(N * HIDDEN + T - 1) / T, T, 0, stream>>>(out1, b1, h1, N);

  // layer 2: h2 = h1b @ W2 ; out = scatter(norm * h2) + b2
  {
    int waves = N / 16;                        // 6250 tiles
    int blocks = (waves + 3) / 4;
    k_gemm2<<<blocks, 128, 0, stream>>>(h1, W2, out1, N);  // out1 reused as h2
  }
  k_self_init16<<<(N * OUT_CH + T - 1) / T, T, 0, stream>>>(dinv, out1, b2, out, N);
  k_scatter16<<<(E * 4 + T - 1) / T, T, 0, stream>>>(src, dst, ew, dinv, out1, out, E);
}